// VoxelToPoint_71116068488049
// MI455X (gfx1250) — compile-verified
//
#include <hip/hip_runtime.h>

// VoxelToPoint gather for MI455X (gfx1250, wave32).
//
// Layout facts (from reference): vf is (B, C=64, D=64, H=64, W=64) f32,
// coords is (B, N, 3) i32 (x,y,z), out is (B, N, 64) f32.
// flat spatial idx = (x*64 + y)*64 + z ; channel stride = 64*64*64 = 262144 floats.
//
// Mapping: 16 threads per point; thread t -> point n = t/16, channels
// c0 = (t%16)*4 .. c0+3.  Each thread: 4 strided 4B gathers (served by L2,
// one 64MB batch volume resident at a time) + one 16B coalesced NT store.

#define C_CH   64
#define DHW    (64 * 64 * 64)   // 262144

// Native clang vector type: accepted by __builtin_nontemporal_store
// (HIP's float4 struct is not).
typedef __attribute__((ext_vector_type(4))) float v4f;

__global__ __launch_bounds__(256) void voxel_gather_kernel(
    const float* __restrict__ vf,     // (B, 64, 262144)
    const int*   __restrict__ coords, // (B, N, 3)
    float*       __restrict__ out,    // (B, N, 64)
    int N)
{
    const int t = blockIdx.x * blockDim.x + threadIdx.x;
    const int n = t >> 4;                 // point index within batch
    if (n >= N) return;

    const int b  = blockIdx.y;            // batch (slow grid dim -> batch-serial, L2-friendly)
    const int c0 = (t & 15) << 2;         // first of 4 channels handled by this thread

    // Coordinates: stream-once -> non-temporal loads (don't pollute L2).
    const int* cp = coords + ((size_t)b * (size_t)N + (size_t)n) * 3;
    const int x = __builtin_nontemporal_load(cp + 0);
    const int y = __builtin_nontemporal_load(cp + 1);
    const int z = __builtin_nontemporal_load(cp + 2);

    // Reference semantics: valid iff upper-bound check passes; clip for safe gather.
    const bool valid = (x < 64) & (y < 64) & (z < 64);
    const int xi = min(max(x, 0), 63);
    const int yi = min(max(y, 0), 63);
    const int zi = min(max(z, 0), 63);
    const int idx = (xi * 64 + yi) * 64 + zi;

    // Gather 4 channels, 1MB apart each (RT temporal hint: we WANT L2 reuse here).
    const float* base = vf + (size_t)b * (size_t)(C_CH * DHW)
                           + (size_t)c0 * (size_t)DHW
                           + (size_t)idx;
    v4f v = (v4f)(0.0f);
    if (valid) {
        v.x = base[0 * DHW];
        v.y = base[1 * DHW];
        v.z = base[2 * DHW];
        v.w = base[3 * DHW];
    }

    // Output: stream-once, perfectly coalesced 128-bit non-temporal store.
    float* op = out + ((size_t)b * (size_t)N + (size_t)n) * C_CH + c0;
    __builtin_nontemporal_store(v, reinterpret_cast<v4f*>(op));
}

extern "C" void kernel_launch(void* const* d_in, const int* in_sizes, int n_in,
                              void* d_out, int out_size, void* d_ws, size_t ws_size,
                              hipStream_t stream)
{
    const float* vf     = (const float*)d_in[0];
    const int*   coords = (const int*)d_in[1];
    float*       out    = (float*)d_out;

    // Derive B and N from input sizes (C, D, H, W fixed at 64 by the reference).
    const int B  = in_sizes[0] / (C_CH * DHW);          // = 4
    const int BN = in_sizes[1] / 3;                     // = B * N
    const int N  = (B > 0) ? (BN / B) : 0;              // = 100000
    if (B <= 0 || N <= 0) return;

    const int threads_per_batch = N * 16;               // 16 threads per point
    dim3 block(256, 1, 1);
    dim3 grid((threads_per_batch + 255) / 256, B, 1);   // grid.x fastest -> batches serialize
    voxel_gather_kernel<<<grid, block, 0, stream>>>(vf, coords, out, N);
}